// HungarianMatcher_90804198572657
// MI455X (gfx1250) — compile-verified
//
#include <hip/hip_runtime.h>
#include <hip/hip_bf16.h>

typedef __attribute__((ext_vector_type(2))) float v2f;
typedef __attribute__((ext_vector_type(8))) float v8f;

namespace {
constexpr int   kNC     = 2;      // classes
constexpr int   kT      = 64;     // targets
constexpr int   kD      = 63;     // keypoint coords
constexpr int   kQ      = 16 * 1024;
constexpr int   kQBlk   = 64;     // queries per workgroup (4 waves x 16 rows)
constexpr int   kThr    = 128;    // 4 wave32 waves
constexpr float kAlpha  = 0.25f;
constexpr float kEps    = 1e-8f;
constexpr float kNegBig = -1e30f; // occluded-coordinate sentinel (kills max() term)
}

// C[q,t] = 2*sum_d max(a_qd, btilde_td) - (A*W^T)[q,t] - r[t] + cost_class[q,t]
//   btilde = b (visible) | -1e30 (occluded);  W[t,d] = 1 (visible) | 2 (occluded) | 0 (pad)
//   r[t] = sum_d m*b.   Exactly equals sum_d m*|a-b| + cost_class.
__global__ __launch_bounds__(kThr)
void matcher_cost_kernel(const float* __restrict__ logits,
                         const float* __restrict__ kp,
                         const float* __restrict__ tgt_kp,
                         const int*  __restrict__ tgt_ids,
                         float* __restrict__ out)
{
  __shared__ __attribute__((aligned(16))) float Abuf[kQBlk * kD + 4]; // verbatim kp tile, stride 63
  __shared__ float Bt[kD][kT];       // masked targets, transposed: bank = t (conflict-free)
  __shared__ __attribute__((aligned(8))) float Wp[32][kT][2];
  // Wp[k>>1][t][k&1]: GEMM weight pairs (k,k+1) contiguous -> single ds_load_b64
  // lands the WMMA B operand in an adjacent VGPR pair (no v_mov shuffles).
  __shared__ float Rs[kT];           // r[t]
  __shared__ float CCs[kNC][kQBlk];  // focal class cost per local query, per class
  __shared__ int   IDs[kT];

  const int tid = threadIdx.x;
  const int qb  = blockIdx.x * kQBlk;

  // ---- Stage 64x63 keypoint tile via CDNA5 async global->LDS DMA (b128 granules).
  {
    unsigned abase = (unsigned)(unsigned long long)(&Abuf[0]);            // low 32 bits = LDS offset
    unsigned long long gsrc = (unsigned long long)(kp + (size_t)qb * kD); // 16B aligned: 64*63*4 | 16
    for (int i = tid; i < (kQBlk * kD) / 4; i += kThr) {                  // 1008 x b128
      unsigned lo = abase + (unsigned)i * 16u;
      unsigned long long ga = gsrc + (unsigned long long)i * 16u;
      asm volatile("global_load_async_to_lds_b128 %0, %1, off"
                   :: "v"(lo), "v"(ga) : "memory");
    }
  }

  // ---- Overlap with DMA: build Bt / Wp / Rs / IDs / focal class costs.
  if (tid < kT) {
    const int t = tid;
    float r = 0.0f;
    for (int d = 0; d < kD; ++d) {
      float b = tgt_kp[t * kD + d];
      bool vis = b > 0.0f;
      Bt[d][t] = vis ? b : kNegBig;
      Wp[d >> 1][t][d & 1] = vis ? 1.0f : 2.0f;
      r += vis ? b : 0.0f;
    }
    Wp[31][t][1] = 0.0f;  // k=63 pad: zero weight kills stride-63 overread
    Rs[t] = r;
    IDs[t] = tgt_ids[t];
  }
  if (tid < kQBlk) {
    const int q = qb + tid;
    #pragma unroll
    for (int c = 0; c < kNC; ++c) {
      float x = logits[q * kNC + c];
      float p = 1.0f / (1.0f + __expf(-x));
      float pos = kAlpha * (1.0f - p) * (1.0f - p) * (-__logf(p + kEps));
      float neg = (1.0f - kAlpha) * p * p * (-__logf(1.0f - p + kEps));
      CCs[c][tid] = pos - neg;
    }
  }
  if (tid == 0) {
    #pragma unroll
    for (int e = 0; e < 4; ++e) Abuf[kQBlk * kD + e] = 0.0f; // tail pad for last row's k=63 read
  }

  asm volatile("s_wait_asynccnt 0x0" ::: "memory");
  __syncthreads();

  const int wave = tid >> 5;   // 0..3: 16-query strip per wave
  const int lane = tid & 31;
  const int tl   = lane & 15;  // N column (t within 16-tile)
  const int hf   = lane >> 4;  // half-wave
  const int qw   = wave * 16;  // local query base

  // ---- G = A * W^T on the matrix pipe: 16 k-steps x 4 t-tiles of v_wmma_f32_16x16x4_f32.
  // A operand: lane M = tl, K = ks*4 + vgpr + 2*hf. B operand: lane N = tl, same K rule.
  v8f G[4] = {};
  const int arow = (qw + tl) * kD;
  #pragma unroll
  for (int ks = 0; ks < 16; ++ks) {
    const int k0 = ks * 4 + 2 * hf;   // even; pair (k0, k0+1) = Wp row k0>>1
    v2f a;
    a[0] = Abuf[arow + k0];
    a[1] = Abuf[arow + k0 + 1];
    #pragma unroll
    for (int tt = 0; tt < 4; ++tt) {
      v2f w = *(const v2f*)(&Wp[k0 >> 1][tt * 16 + tl][0]);  // one ds_load_b64
      G[tt] = __builtin_amdgcn_wmma_f32_16x16x4_f32(
          false, a, false, w, (short)0, G[tt], false, false);
    }
  }

  // ---- S1[tt][j] = sum_d max(a_qd, btilde_td), kept in the WMMA C/D layout:
  // element (vgpr j, lane) = (row M = j + 8*hf, col N = tl). max+add = dual-issue VOP2 pair.
  float S1[4][8];
  #pragma unroll
  for (int tt = 0; tt < 4; ++tt)
    #pragma unroll
    for (int j = 0; j < 8; ++j) S1[tt][j] = 0.0f;

  for (int d = 0; d < kD; ++d) {
    float b0 = Bt[d][tl];        // 16 lanes -> 16 banks, two halves broadcast
    float b1 = Bt[d][16 + tl];
    float b2 = Bt[d][32 + tl];
    float b3 = Bt[d][48 + tl];
    #pragma unroll
    for (int j = 0; j < 8; ++j) {
      float a = Abuf[(qw + j + 8 * hf) * kD + d];  // uniform per half-wave -> LDS broadcast
      S1[0][j] += fmaxf(a, b0);
      S1[1][j] += fmaxf(a, b1);
      S1[2][j] += fmaxf(a, b2);
      S1[3][j] += fmaxf(a, b3);
    }
  }

  // ---- Epilogue: cost = 2*S1 - G - r[t] + cost_class[q, ids[t]].
  #pragma unroll
  for (int tt = 0; tt < 4; ++tt) {
    const int t = tt * 16 + tl;
    const float rv = Rs[t];
    const int id = IDs[t];
    #pragma unroll
    for (int j = 0; j < 8; ++j) {
      const int ql = qw + j + 8 * hf;
      float val = fmaf(2.0f, S1[tt][j], -G[tt][j]) + (CCs[id][ql] - rv);
      out[(size_t)(qb + ql) * kT + t] = val;
    }
  }
}

extern "C" void kernel_launch(void* const* d_in, const int* in_sizes, int n_in,
                              void* d_out, int out_size, void* d_ws, size_t ws_size,
                              hipStream_t stream) {
  (void)in_sizes; (void)n_in; (void)out_size; (void)d_ws; (void)ws_size;
  const float* logits = (const float*)d_in[0];   // [16,1024,2]  f32
  const float* kp     = (const float*)d_in[1];   // [16,1024,63] f32
  const float* tgt    = (const float*)d_in[2];   // [64,63]      f32
  const int*   ids    = (const int*)d_in[3];     // [64]         int
  float*       out    = (float*)d_out;           // [16,1024,64] f32

  dim3 grid(kQ / kQBlk);   // 256 blocks x 128 threads (4 waves) -> 1024 waves
  matcher_cost_kernel<<<grid, dim3(kThr), 0, stream>>>(logits, kp, tgt, ids, out);
}